// MapNet_65867618451748
// MI455X (gfx1250) — compile-verified
//
#include <hip/hip_runtime.h>

// MapNet egocentric projection (scatter-max of image features onto a top-down map)
//
// img_feats : (64, 64, 64, 64)  f32   (bs, F, h, w)
// depth     : (64, 3, 256, 256) f32   (only channel 0 used)
// out       : (64, 64, 128, 128) f32
//
// Memory-bound: ~256MB out + 64MB feats + ~1MB depth => ~14us at 23.3 TB/s.
// Strategy: one workgroup per (batch, channel); the 128x128 map tile lives in
// LDS (64KB) as sign-flip-encoded uint32 so scatter-max is ds_max_u32. The
// tile is decoded in place (EPS -> 0) and then flushed with the CDNA5 async
// LDS->global path (global_store_async_from_lds_b128, ASYNCcnt-tracked),
// avoiding the LDS -> VGPR -> memory round-trip for the dominant 256MB store.

#define MAP_S    128
#define NCELL    (MAP_S * MAP_S)     // 16384 cells
#define FEAT_H   64
#define FEAT_W   64
#define NPIX     (FEAT_H * FEAT_W)   // 4096 pixels
#define FDIM     64
#define BS       64
#define DEPTH_H  256
#define DEPTH_W  256
#define NTHREADS 256

#define AS_GLOBAL __attribute__((address_space(1)))
#define AS_LOCAL  __attribute__((address_space(3)))

// Vector type matching the builtin's expected parameter type:
// 'int __vector_size__(4*sizeof(int))'
typedef __attribute__((__vector_size__(4 * sizeof(int)))) int v4i;

#if defined(__has_builtin)
#  if __has_builtin(__builtin_amdgcn_global_store_async_from_lds_b128)
#    define HAVE_ASYNC_BUILTIN 1
#  endif
#endif
#ifndef HAVE_ASYNC_BUILTIN
#  define HAVE_ASYNC_BUILTIN 0
#endif

// Monotonic order-preserving float->uint encoding:
// negative floats -> ~bits (top bit 0), non-negative -> bits | 0x80000000.
__device__ __forceinline__ unsigned flipf(float f) {
  unsigned u = __float_as_uint(f);
  return (u & 0x80000000u) ? ~u : (u | 0x80000000u);
}
// Inverse, pure-integer form (returns the float's bit pattern).
__device__ __forceinline__ unsigned unflip_bits(unsigned u) {
  return (u & 0x80000000u) ? (u ^ 0x80000000u) : ~u;
}

// One 16-byte lane store, LDS -> global, tracked with ASYNCcnt.
__device__ __forceinline__ void async_store_b128(void* gp, void* lp) {
#if HAVE_ASYNC_BUILTIN
  __builtin_amdgcn_global_store_async_from_lds_b128(
      (AS_GLOBAL v4i*)gp, (AS_LOCAL v4i*)lp, /*offset=*/0, /*cpol=*/0);
#else
  // GV mode: VADDR = 64-bit global address pair, VSRC = LDS byte offset.
  // Low 32 bits of a flat LDS pointer are the DS byte offset (aperture rule).
  unsigned long long ga = (unsigned long long)gp;
  unsigned la = (unsigned)(unsigned long long)lp;
  asm volatile("global_store_async_from_lds_b128 %0, %1, off"
               :: "v"(ga), "v"(la) : "memory");
#endif
}

__device__ __forceinline__ void wait_asynccnt0() {
#if defined(__has_builtin) && __has_builtin(__builtin_amdgcn_s_wait_asynccnt)
  __builtin_amdgcn_s_wait_asynccnt(0);
#else
  asm volatile("s_wait_asynccnt 0x0" ::: "memory");
#endif
}

__global__ __launch_bounds__(NTHREADS)
void mapnet_scatter_kernel(const float* __restrict__ img_feats,
                           const float* __restrict__ depth,
                           float* __restrict__ out) {
  __shared__ unsigned smem[NCELL];   // 64 KB map tile (encoded floats)

  const int f = blockIdx.x;          // channel
  const int b = blockIdx.y;          // batch
  const int t = threadIdx.x;

  const unsigned ENC_EPS = flipf(-1e16f);   // encoded EPS (compile-time const)

  // ---- init map tile to encoded EPS (uint4 => ds_store_b128) ----
  uint4* s4 = (uint4*)smem;
#pragma unroll
  for (int i = 0; i < NCELL / 4 / NTHREADS; ++i) {  // 16 iters
    s4[i * NTHREADS + t] = make_uint4(ENC_EPS, ENC_EPS, ENC_EPS, ENC_EPS);
  }
  __syncthreads();

  const float* feat_bf = img_feats + (size_t)(b * FDIM + f) * NPIX;
  const float* depth_b = depth + (size_t)b * 3 * DEPTH_H * DEPTH_W;  // channel 0

  // ---- scatter-max: 4096 pixels, 256 threads => 16 iterations ----
#pragma unroll 1
  for (int base = 0; base < NPIX; base += NTHREADS) {
    const int p = base + t;
    // prefetch next feature chunk (gfx1250 global_prefetch_b8; speculative OK)
    __builtin_prefetch(feat_bf + p + NTHREADS, 0, 1);

    const int i   = p >> 6;            // feature row
    const int j   = p & 63;            // feature col
    const int row = i * 4 + 2;         // K=4, sample at pixel centers
    const int col = j * 4 + 2;

    const float d = depth_b[row * DEPTH_W + col];
    const float Z = d * 10.0f;                                    // meters
    const float X = ((float)col - 128.0f) * (1.0f / 128.0f) * Z;  // (x-cx)/fx * Z

    const bool valid = fabsf(Z) > 0.8f;
    // jnp.round == round-half-to-even == rintf under default RNE mode
    const int zi = (int)rintf(-(Z / 0.1f) + 127.0f);
    const int xi = (int)rintf(X / 0.1f + 63.5f);

    const bool oob = (zi >= MAP_S) | (xi >= MAP_S) | (zi < 0) | (xi < 0) | (!valid);
    // Reference writes EPS to cell 0 for invalid pixels: a no-op under max()
    // against an EPS-initialized grid, so we simply skip them.
    if (!oob) {
      const float v = feat_bf[p];
      atomicMax(&smem[zi * MAP_S + xi], flipf(v));   // ds_max_u32
    }
  }
  __syncthreads();

  // ---- decode in place: encoded uint -> final output bits (EPS -> 0) ----
#pragma unroll 1
  for (int c = 0; c < NCELL / 4 / NTHREADS; ++c) {  // 16 iters
    const uint4 u = s4[c * NTHREADS + t];           // ds_load_b128
    uint4 o;
    o.x = (u.x == ENC_EPS) ? 0u : unflip_bits(u.x);
    o.y = (u.y == ENC_EPS) ? 0u : unflip_bits(u.y);
    o.z = (u.z == ENC_EPS) ? 0u : unflip_bits(u.z);
    o.w = (u.w == ENC_EPS) ? 0u : unflip_bits(u.w);
    s4[c * NTHREADS + t] = o;                       // ds_store_b128
  }
  __syncthreads();  // all LDS writes visible before async engine reads them

  // ---- async flush: LDS -> global, no VGPR round-trip ----
  float4* out4 = (float4*)(out + (size_t)(b * FDIM + f) * NCELL);
#pragma unroll 1
  for (int c = 0; c < NCELL / 4 / NTHREADS; ++c) {  // 16 iters x 16B/lane
    async_store_b128(out4 + c * NTHREADS + t, &s4[c * NTHREADS + t]);
  }
  wait_asynccnt0();   // s_wait_asynccnt 0
}

extern "C" void kernel_launch(void* const* d_in, const int* in_sizes, int n_in,
                              void* d_out, int out_size, void* d_ws, size_t ws_size,
                              hipStream_t stream) {
  const float* img_feats = (const float*)d_in[0];   // (64,64,64,64)
  const float* depth     = (const float*)d_in[1];   // (64,3,256,256)
  float* out             = (float*)d_out;           // (64,64,128,128)

  dim3 grid(FDIM, BS);   // one workgroup per (channel, batch)
  dim3 block(NTHREADS);
  mapnet_scatter_kernel<<<grid, block, 0, stream>>>(img_feats, depth, out);
}